// Vortex_Particles_37830071943726
// MI455X (gfx1250) — compile-verified
//
#include <hip/hip_runtime.h>

typedef __attribute__((ext_vector_type(2))) float v2f;
typedef __attribute__((ext_vector_type(8))) float v8f;

#define PARTICLES 256
#define PT_TILES  16      // 256 particles / 16 per tile
#define A_STRIDE  20      // padded LDS stride (floats) -> conflict-free b64 loads

// ---------------------------------------------------------------------------
// Setup: per-particle WMMA A-rows + scalar constants into workspace.
//   A_d2 = [-2p.x, -2p.y, -2p.z, |p|^2]            (dist^2 via B=[c,1], C=|c|^2)
//   A_cx = [0, -wd.z,  wd.y, p.y*wd.z - p.z*wd.y]  (crossX of (p-c) x wd)
//   A_cy = [ wd.z, 0, -wd.x, p.z*wd.x - p.x*wd.z]
//   A_cz = [-wd.y,  wd.x, 0, p.x*wd.y - p.y*wd.x]
//   e1 = -log2(e)/(2r^2), c0 = 1/(r^3*40000), lim = 9r^2, wd = w*dir
// ---------------------------------------------------------------------------
__global__ void vp_setup_kernel(const float* __restrict__ ppos,
                                const float* __restrict__ pdir,
                                const float* __restrict__ pint,
                                const float* __restrict__ raw,
                                const float* __restrict__ ru,
                                const int*   __restrict__ tmask,
                                const int*   __restrict__ tptr,
                                int P, int T,
                                float* __restrict__ pA,
                                float* __restrict__ e1,
                                float* __restrict__ c0,
                                float* __restrict__ lim)
{
    int p = blockIdx.x * blockDim.x + threadIdx.x;
    if (p >= P) return;
    int ti = tptr[0];

    float px = ppos[(p*T + ti)*3 + 0];
    float py = ppos[(p*T + ti)*3 + 1];
    float pz = ppos[(p*T + ti)*3 + 2];
    float dx = pdir[(p*T + ti)*3 + 0];
    float dy = pdir[(p*T + ti)*3 + 1];
    float dz = pdir[(p*T + ti)*3 + 2];
    float mk = (float)tmask[p*T + ti];

    float w = sqrtf(fminf(fmaxf(raw[p], 0.0f), 10.0f) + 1e-8f) * pint[p*T + ti] * mk;
    float r = fmaxf(0.0f, 0.2f * (0.5f * ru[p] + 1.0f));
    float r2 = r * r;

    e1[p]  = -1.4426950408889634f / (2.0f * r2);   // exp(-x) = exp2(x * e1 / ...)
    c0[p]  = 1.0f / (r2 * r * 40000.0f);
    lim[p] = 9.0f * r2;

    float wx = w * dx, wy = w * dy, wz = w * dz;

    float* A = pA + p * 16;
    A[0]  = -2.0f*px; A[1]  = -2.0f*py; A[2]  = -2.0f*pz; A[3]  = px*px + py*py + pz*pz;
    A[4]  = 0.0f;     A[5]  = -wz;      A[6]  =  wy;      A[7]  = py*wz - pz*wy;
    A[8]  =  wz;      A[9]  = 0.0f;     A[10] = -wx;      A[11] = pz*wx - px*wz;
    A[12] = -wy;      A[13] =  wx;      A[14] = 0.0f;     A[15] = px*wy - py*wx;
}

// ---------------------------------------------------------------------------
// Main: one wave32 per 16-coord tile; loop over 16 particle tiles.
// 4 x v_wmma_f32_16x16x4_f32 per tile give dist^2 and crossX/Y/Z; VALU does
// exp2/rsq/mask/accumulate. D layout: VGPR v -> particle m = v + 8*hi,
// coord n = lane%16; lanes l and l+16 share coord -> one shfl_xor(16) reduce.
// ---------------------------------------------------------------------------
__global__ void __launch_bounds__(256)
vp_main_kernel(const float* __restrict__ coord,
               const float* __restrict__ pA,
               const float* __restrict__ e1,
               const float* __restrict__ c0,
               const float* __restrict__ lim,
               float* __restrict__ out,
               int N, int nTiles)
{
    __shared__ __align__(32) float sA[PARTICLES * A_STRIDE];
    __shared__ __align__(32) float sE[PARTICLES];
    __shared__ __align__(32) float sC[PARTICLES];
    __shared__ __align__(32) float sL[PARTICLES];

    int tid = threadIdx.x;
    for (int i = tid; i < PARTICLES * 16; i += blockDim.x) {
        int p = i >> 4, j = i & 15;
        sA[p * A_STRIDE + j] = pA[i];
    }
    for (int i = tid; i < PARTICLES; i += blockDim.x) {
        sE[i] = e1[i]; sC[i] = c0[i]; sL[i] = lim[i];
    }
    __syncthreads();

    int wave = tid >> 5;
    int lane = tid & 31;
    int tile = blockIdx.x * (blockDim.x >> 5) + wave;
    if (tile >= nTiles) return;            // wave-uniform: EXEC stays all-ones

    int n  = lane & 15;
    int hi = lane >> 4;                    // K-half select for A/B operands
    int ci = tile * 16 + n;
    int cl = ci < N ? ci : N - 1;

    float cx = coord[cl*3 + 0];
    float cy = coord[cl*3 + 1];
    float cz = coord[cl*3 + 2];

    // B (4x16): VGPR0 = K0/K2 = c.x/c.z ; VGPR1 = K1/K3 = c.y/1
    v2f b;
    b.x = hi ? cz : cx;
    b.y = hi ? 1.0f : cy;

    float cn2 = cx*cx + cy*cy + cz*cz;
    v8f cd2, vz;
#pragma unroll
    for (int i = 0; i < 8; i++) { cd2[i] = cn2; vz[i] = 0.0f; }

    float accx = 0.0f, accy = 0.0f, accz = 0.0f;

    for (int t = 0; t < PT_TILES; t++) {
        const float* ab = &sA[(t * 16 + n) * A_STRIDE];
        v2f a0 = *(const v2f*)(ab + 0  + 2*hi);   // dist^2 row
        v2f a1 = *(const v2f*)(ab + 4  + 2*hi);   // crossX row
        v2f a2 = *(const v2f*)(ab + 8  + 2*hi);   // crossY row
        v2f a3 = *(const v2f*)(ab + 12 + 2*hi);   // crossZ row

        int sb = t * 16 + 8 * hi;                  // my 8 particles (m = v + 8*hi)
        v8f e1v = *(const v8f*)&sE[sb];
        v8f c0v = *(const v8f*)&sC[sb];
        v8f lmv = *(const v8f*)&sL[sb];

        v8f d2 = __builtin_amdgcn_wmma_f32_16x16x4_f32(false, a0, false, b, (short)0, cd2, false, false);
        v8f cX = __builtin_amdgcn_wmma_f32_16x16x4_f32(false, a1, false, b, (short)0, vz,  false, false);
        v8f cY = __builtin_amdgcn_wmma_f32_16x16x4_f32(false, a2, false, b, (short)0, vz,  false, false);
        v8f cZ = __builtin_amdgcn_wmma_f32_16x16x4_f32(false, a3, false, b, (short)0, vz,  false, false);

#pragma unroll
        for (int i = 0; i < 8; i++) {
            float q = fmaxf(d2[i], 1e-12f);
            float s = __builtin_amdgcn_exp2f(q * e1v[i]) * c0v[i] * __builtin_amdgcn_rsqf(q);
            s = (q < lmv[i]) ? s : 0.0f;
            accx = fmaf(s, cX[i], accx);
            accy = fmaf(s, cY[i], accy);
            accz = fmaf(s, cZ[i], accz);
        }
    }

    // lanes l and l+16 hold partial sums (particles m=0..7 vs 8..15) for the
    // same coord n: one cross-half add completes the particle reduction.
    accx += __shfl_xor(accx, 16, 32);
    accy += __shfl_xor(accy, 16, 32);
    accz += __shfl_xor(accz, 16, 32);

    if (hi == 0 && ci < N) {
        out[ci*3 + 0] = accx;
        out[ci*3 + 1] = accy;
        out[ci*3 + 2] = accz;
    }
}

// ---------------------------------------------------------------------------
extern "C" void kernel_launch(void* const* d_in, const int* in_sizes, int n_in,
                              void* d_out, int out_size, void* d_ws, size_t ws_size,
                              hipStream_t stream)
{
    (void)n_in; (void)out_size; (void)ws_size;

    const float* coord = (const float*)d_in[0];
    const float* ppos  = (const float*)d_in[1];
    const float* pdir  = (const float*)d_in[2];
    const float* pint  = (const float*)d_in[3];
    const float* raw   = (const float*)d_in[4];
    const float* ru    = (const float*)d_in[5];
    const int*   tmask = (const int*)d_in[6];
    const int*   tptr  = (const int*)d_in[7];

    int N = in_sizes[0] / 3;          // 262144 coords
    int P = in_sizes[4];              // 256 particles
    int T = in_sizes[6] / P;          // 8 time steps

    float* pA  = (float*)d_ws;        // P*16 floats of WMMA A rows
    float* e1  = pA + (size_t)P * 16;
    float* c0  = e1 + P;
    float* lim = c0 + P;

    vp_setup_kernel<<<(P + 255) / 256, 256, 0, stream>>>(
        ppos, pdir, pint, raw, ru, tmask, tptr, P, T, pA, e1, c0, lim);

    int nTiles = (N + 15) / 16;
    int wavesPerBlock = 256 / 32;
    int blocks = (nTiles + wavesPerBlock - 1) / wavesPerBlock;
    vp_main_kernel<<<blocks, 256, 0, stream>>>(
        coord, pA, e1, c0, lim, (float*)d_out, N, nTiles);
}